// GTLayer_21105469292738
// MI455X (gfx1250) — compile-verified
//
#include <hip/hip_runtime.h>
#include <cstddef>

typedef float v2f __attribute__((ext_vector_type(2)));
typedef float v4f __attribute__((ext_vector_type(4)));
typedef float v8f __attribute__((ext_vector_type(8)));

#define DMODEL 96
#define NHEADS 4
#define DHEAD  24

// ---------------------------------------------------------------------------
// Kernel 0: residual init (out = embeds) + zero the softmax normalizers.
// Required every call: harness poisons d_out/d_ws and we accumulate into out.
// ---------------------------------------------------------------------------
__global__ __launch_bounds__(256) void gt_init(const float* __restrict__ embeds,
                                               float* __restrict__ out,
                                               float* __restrict__ norms,
                                               int nNodes) {
  int i = blockIdx.x * 256 + threadIdx.x;
  int total = nNodes * DMODEL;
  if (i < total) out[i] = embeds[i];
  if (i < nNodes * NHEADS) norms[i] = 0.0f;
}

// ---------------------------------------------------------------------------
// Kernel 1: node-level projections via V_WMMA_F32_16X16X4_F32.
//   Q = embeds @ qW, K = embeds @ kW, V = embeds @ vW   (all [N,96]@[96,96])
// One wave computes one 16x16 output tile; 24 WMMA steps over K=96.
// Tile guard is wave-uniform -> EXEC all-1s inside active waves (WMMA req).
// A-layout (16x4 f32): lane l -> row (l&15), K = 2*(l>>4) + {0,1} in {v0,v1}.
// B-layout (4x16 f32): lane l -> col (l&15), K = 2*(l>>4) + {0,1}.
// C/D-layout: VGPR g -> row g + 8*(l>>4), col (l&15).
// ---------------------------------------------------------------------------
__global__ __launch_bounds__(256) void gt_proj_wmma(
    const float* __restrict__ embeds,
    const float* __restrict__ qW, const float* __restrict__ kW,
    const float* __restrict__ vW,
    float* __restrict__ Q, float* __restrict__ K, float* __restrict__ V,
    int nMTiles) {
  const int lane = threadIdx.x & 31;
  const int wave = threadIdx.x >> 5;
  const int tile = blockIdx.x * 8 + wave;
  const int nTiles = nMTiles * 18;          // 6 n-tiles x 3 matrices per m-tile
  if (tile >= nTiles) return;               // wave-uniform

  const int mt  = tile / 18;
  const int sub = tile % 18;
  const int mat = sub / 6;                  // 0=Q, 1=K, 2=V
  const int n0  = (sub % 6) * 16;
  const int m0  = mt * 16;

  const float* W   = (mat == 0) ? qW : (mat == 1) ? kW : vW;
  float*       Out = (mat == 0) ? Q  : (mat == 1) ? K  : V;

  const int halfSel = lane >> 4;            // 0: K pair {0,1}; 1: K pair {2,3}
  const int l15     = lane & 15;
  const int kOff    = halfSel * 2;

  const float* arow = embeds + (size_t)(m0 + l15) * DMODEL;   // A row per lane
  const float* bcol = W + n0 + l15;                           // B column per lane

  v8f c = {0.f, 0.f, 0.f, 0.f, 0.f, 0.f, 0.f, 0.f};
#pragma unroll
  for (int kk = 0; kk < DMODEL; kk += 4) {
    v2f a = *(const v2f*)(arow + kk + kOff);          // 8B aligned (kk+kOff even)
    v2f b;
    b.x = bcol[(size_t)(kk + kOff)     * DMODEL];
    b.y = bcol[(size_t)(kk + kOff + 1) * DMODEL];
    // D = A(16x4) * B(4x16) + C ; full fp32, round-to-nearest-even
    c = __builtin_amdgcn_wmma_f32_16x16x4_f32(false, a, false, b,
                                              (short)0, c, false, false);
  }

  float* orow = Out + (size_t)m0 * DMODEL + n0 + l15;
#pragma unroll
  for (int g = 0; g < 8; ++g)
    orow[(size_t)(g + halfSel * 8) * DMODEL] = c[g];
}

// ---------------------------------------------------------------------------
// Kernel 2: per-(edge,head) score = clamp(dot(Q[r,h,:], K[c,h,:]), +-10);
// exp_s stored per edge-head; atomic accumulate normalizer per (row, head).
// 4 threads of one edge cover one contiguous 384B row -> coalesced b128 loads.
// Gathers hit L2 (Q/K arrays ~19MB each, L2 = 192MB).
// ---------------------------------------------------------------------------
__global__ __launch_bounds__(256) void gt_edge_scores(
    const float* __restrict__ Q, const float* __restrict__ K,
    const int* __restrict__ rows, const int* __restrict__ cols,
    float* __restrict__ exp_s, float* __restrict__ norms, int nEdges) {
  int tid = blockIdx.x * 256 + threadIdx.x;
  if (tid >= nEdges * NHEADS) return;
  int e = tid >> 2, h = tid & 3;
  int r = rows[e], c = cols[e];
  const v4f* q = (const v4f*)(Q + (size_t)r * DMODEL + h * DHEAD);  // 16B aligned
  const v4f* k = (const v4f*)(K + (size_t)c * DMODEL + h * DHEAD);
  float s = 0.0f;
#pragma unroll
  for (int j = 0; j < 6; ++j) {
    v4f a = q[j], b = k[j];
    s += a.x * b.x + a.y * b.y + a.z * b.z + a.w * b.w;
  }
  s = fminf(10.0f, fmaxf(-10.0f, s));
  float es = expf(s);
  exp_s[tid] = es;
  atomicAdd(&norms[(size_t)r * NHEADS + h], es);
}

// ---------------------------------------------------------------------------
// Kernel 3: per-(edge,head) weighted scatter: out[r,h,:] += w * V[c,h,:].
// 24 global_atomic_add_f32 per thread, all L2-resident.
// ---------------------------------------------------------------------------
__global__ __launch_bounds__(256) void gt_edge_aggregate(
    const float* __restrict__ V,
    const int* __restrict__ rows, const int* __restrict__ cols,
    const float* __restrict__ exp_s, const float* __restrict__ norms,
    float* __restrict__ out, int nEdges) {
  int tid = blockIdx.x * 256 + threadIdx.x;
  if (tid >= nEdges * NHEADS) return;
  int e = tid >> 2, h = tid & 3;
  int r = rows[e], c = cols[e];
  float w = exp_s[tid] / (norms[(size_t)r * NHEADS + h] + 1e-8f);
  const v4f* v = (const v4f*)(V + (size_t)c * DMODEL + h * DHEAD);
  float* o = out + (size_t)r * DMODEL + h * DHEAD;
#pragma unroll
  for (int j = 0; j < 6; ++j) {
    v4f t = v[j];
    atomicAdd(o + 4 * j + 0, w * t.x);
    atomicAdd(o + 4 * j + 1, w * t.y);
    atomicAdd(o + 4 * j + 2, w * t.z);
    atomicAdd(o + 4 * j + 3, w * t.w);
  }
}

// ---------------------------------------------------------------------------
// Launch: init -> WMMA projections -> scores/normalizers -> aggregation.
// Stream ordering provides all inter-kernel dependencies; no sync calls
// (graph-capture safe). Workspace: Q|K|V|exp_s|norms = ~71.2 MB of d_ws.
// ---------------------------------------------------------------------------
extern "C" void kernel_launch(void* const* d_in, const int* in_sizes, int n_in,
                              void* d_out, int out_size, void* d_ws,
                              size_t ws_size, hipStream_t stream) {
  const float* embeds = (const float*)d_in[0];
  const float* qW     = (const float*)d_in[1];
  const float* kW     = (const float*)d_in[2];
  const float* vW     = (const float*)d_in[3];
  const int*   rows   = (const int*)d_in[4];
  const int*   cols   = (const int*)d_in[5];
  float*       out    = (float*)d_out;

  const int nNodes = in_sizes[0] / DMODEL;   // 50000
  const int nEdges = in_sizes[4];            // 800000

  float* ws = (float*)d_ws;
  const size_t nd = (size_t)nNodes * DMODEL;
  float* Q     = ws;
  float* Kp    = Q + nd;
  float* Vp    = Kp + nd;
  float* exp_s = Vp + nd;
  float* norms = exp_s + (size_t)nEdges * NHEADS;

  {
    int total = nNodes * DMODEL;
    gt_init<<<(total + 255) / 256, 256, 0, stream>>>(embeds, out, norms, nNodes);
  }
  {
    int nMTiles = nNodes / 16;               // 50000 is an exact multiple of 16
    int nTiles  = nMTiles * 18;
    int blocks  = (nTiles + 7) / 8;          // 8 waves (tiles) per 256-thread block
    gt_proj_wmma<<<blocks, 256, 0, stream>>>(embeds, qW, kW, vW, Q, Kp, Vp,
                                             nMTiles);
  }
  {
    int total = nEdges * NHEADS;
    gt_edge_scores<<<(total + 255) / 256, 256, 0, stream>>>(Q, Kp, rows, cols,
                                                            exp_s, norms, nEdges);
    gt_edge_aggregate<<<(total + 255) / 256, 256, 0, stream>>>(Vp, rows, cols,
                                                               exp_s, norms, out,
                                                               nEdges);
  }
}